// cbowns_1125281432287
// MI455X (gfx1250) — compile-verified
//
#include <hip/hip_runtime.h>
#include <hip/hip_bf16.h>

// CBOW negative-sampling loss, WMMA formulation for gfx1250 (wave32).
//
// Per wave: 16 items. A-matrix = target rows (16xE), B-matrix = ctx-mean /
// neg-sum rows (Ex16 fed 4 K at a time). D = A*B accumulated over E; we use
// only the diagonal D[m][m] = dot(tgt_m, reduced_m). f32 WMMA 16x16x4 keeps
// full f32 precision and its A/B lane layouts are identical (row = lane%16,
// k-sub = (lane/16)*2), so every gathered element is loaded exactly once.

typedef __attribute__((ext_vector_type(2))) float v2f;
typedef __attribute__((ext_vector_type(8))) float v8f;

#define EMB 128
#define WIN 8
#define NEGK 3
#define WAVES 8
#define TILE 16
#define ITEMS_PER_BLOCK (WAVES * TILE)

__device__ __forceinline__ float log_sigmoid(float x) {
    // stable: min(x,0) - log1p(exp(-|x|))
    return fminf(x, 0.0f) - log1pf(expf(-fabsf(x)));
}

__global__ __launch_bounds__(WAVES * 32) void cbow_neg_wmma_kernel(
    const int* __restrict__ targets,
    const int* __restrict__ contexts,
    const int* __restrict__ negsamples,
    const float* __restrict__ target_W,
    const float* __restrict__ context_W,
    float* __restrict__ partials, int B)
{
    __shared__ float sdiag[WAVES][2][256];  // 16x16 D tiles (pos, neg)
    __shared__ float sred[WAVES];

    const int lane = threadIdx.x & 31;
    const int wave = threadIdx.x >> 5;
    const int m    = lane & 15;          // matrix row this lane serves
    const int kh   = (lane >> 4) << 1;   // k sub-offset within 4-chunk: 0 or 2

    int item = blockIdx.x * ITEMS_PER_BLOCK + wave * TILE + m;
    const bool valid = (item < B);
    if (!valid) item = B - 1;            // clamp; contribution zeroed later

    // Per-lane row base offsets (in floats). Lanes L and L+16 share indices
    // but read different k-halves of each 4-wide chunk.
    const long toff = (long)targets[item] * EMB + kh;
    long coff[WIN];
#pragma unroll
    for (int w = 0; w < WIN; ++w)
        coff[w] = (long)contexts[item * WIN + w] * EMB + kh;
    long noff[NEGK];
#pragma unroll
    for (int k = 0; k < NEGK; ++k)
        noff[k] = (long)negsamples[item * NEGK + k] * EMB + kh;

    v8f accP = {};   // D[m][n] = dot(tgt_m, ctx_mean_n)
    v8f accN = {};   // D[m][n] = dot(tgt_m, neg_sum_n)

#pragma unroll 2
    for (int k0 = 0; k0 < EMB; k0 += 4) {
        // A operand: 2 consecutive f32 of target row (this lane's k-half)
        v2f a = *(const v2f*)(target_W + toff + k0);

        // B operand for pos: on-the-fly context mean
        v2f bc = {0.0f, 0.0f};
#pragma unroll
        for (int w = 0; w < WIN; ++w)
            bc += *(const v2f*)(context_W + coff[w] + k0);
        bc *= (1.0f / (float)WIN);

        // B operand for neg: sum of negative rows
        v2f bn = {0.0f, 0.0f};
#pragma unroll
        for (int k = 0; k < NEGK; ++k)
            bn += *(const v2f*)(context_W + noff[k] + k0);

        // 8 args: (neg_a, A, neg_b, B, c_mod, C, reuse_a, reuse_b)
        accP = __builtin_amdgcn_wmma_f32_16x16x4_f32(
            false, a, false, bc, (short)0, accP, false, false);
        accN = __builtin_amdgcn_wmma_f32_16x16x4_f32(
            false, a, false, bn, (short)0, accN, false, false);
    }

    // Scatter D to LDS per the 16x16 f32 C/D layout:
    // VGPR r: lanes 0-15 -> (M=r, N=lane), lanes 16-31 -> (M=r+8, N=lane-16)
    {
        const int Mbase = (lane >> 4) << 3;  // 0 or 8
#pragma unroll
        for (int r = 0; r < 8; ++r) {
            sdiag[wave][0][(Mbase + r) * 16 + m] = accP[r];
            sdiag[wave][1][(Mbase + r) * 16 + m] = accN[r];
        }
    }
    __syncthreads();

    float v = 0.0f;
    if (lane < 16 && valid) {
        float pos =  sdiag[wave][0][lane * 16 + lane];
        float neg = -sdiag[wave][1][lane * 16 + lane];
        v = log_sigmoid(pos) + log_sigmoid(neg);
    }

    // wave32 reduction (lanes 16-31 carry zero)
#pragma unroll
    for (int off = 16; off; off >>= 1)
        v += __shfl_xor(v, off, 32);
    if (lane == 0) sred[wave] = v;
    __syncthreads();

    if (threadIdx.x == 0) {
        float s = 0.0f;
#pragma unroll
        for (int w = 0; w < WAVES; ++w) s += sred[w];
        partials[blockIdx.x] = s;
    }
}

__global__ __launch_bounds__(256) void cbow_reduce_kernel(
    const float* __restrict__ partials, int n,
    float* __restrict__ out, float inv_b)
{
    __shared__ float sm[256];
    float s = 0.0f;
    for (int i = threadIdx.x; i < n; i += 256) s += partials[i];
    sm[threadIdx.x] = s;
    __syncthreads();
    for (int step = 128; step; step >>= 1) {
        if ((int)threadIdx.x < step) sm[threadIdx.x] += sm[threadIdx.x + step];
        __syncthreads();
    }
    if (threadIdx.x == 0) out[0] = -sm[0] * inv_b;   // -mean(per_item)
}

extern "C" void kernel_launch(void* const* d_in, const int* in_sizes, int n_in,
                              void* d_out, int out_size, void* d_ws, size_t ws_size,
                              hipStream_t stream)
{
    const int*   targets    = (const int*)d_in[0];
    const int*   contexts   = (const int*)d_in[1];
    const int*   negsamples = (const int*)d_in[2];
    const float* target_W   = (const float*)d_in[3];
    const float* context_W  = (const float*)d_in[4];
    float*       out        = (float*)d_out;

    const int B = in_sizes[0];
    const int nblocks = (B + ITEMS_PER_BLOCK - 1) / ITEMS_PER_BLOCK;  // 2048 for B=262144
    float* partials = (float*)d_ws;  // nblocks floats (8 KB) — well under ws_size

    cbow_neg_wmma_kernel<<<nblocks, WAVES * 32, 0, stream>>>(
        targets, contexts, negsamples, target_W, context_W, partials, B);
    cbow_reduce_kernel<<<1, 256, 0, stream>>>(partials, nblocks, out, 1.0f / (float)B);
}